// SpatiallyVariantConvolutionStridded_10316511445547
// MI455X (gfx1250) — compile-verified
//
#include <hip/hip_runtime.h>

typedef __attribute__((ext_vector_type(2))) float v2f;
typedef __attribute__((ext_vector_type(8))) float v8f;

#define BATCH 8
#define CH    128
#define IMH   256
#define IMW   256
#define KSZ   5
#define PADR  2
#define CT    16                 // channels per tile (WMMA M)
#define WT    16                 // pixels per tile  (WMMA N)
#define HT    4                  // output rows per block
#define FR    (HT + KSZ - 1)     // 8 staged feature rows
#define MW    (WT + KSZ - 1)     // 20 feature columns per tile (banded K)
#define HW    (IMH * IMW)

__global__ __launch_bounds__(32)
void svconv_wmma_kernel(const float* __restrict__ kern,
                        const float* __restrict__ feat,
                        float* __restrict__ out)
{
    // sF[r][c][m]: feature rows h0-2 .. h0+5, 16 channels, 20 columns (zero-padded)
    __shared__ __align__(16) float sF[FR * CT * MW];      // 2560 f32
    // sK[hr][k*5+l][j]: per-pixel kernels for the 4 output rows
    __shared__ __align__(16) float sK[HT * KSZ * KSZ * WT]; // 1600 f32

    const int lane  = threadIdx.x;          // one wave32 per block
    const int w0    = blockIdx.x * WT;
    const int h0    = blockIdx.y * HT;
    const int bz    = blockIdx.z;
    const int ctile = bz & 7;
    const int b     = bz >> 3;
    const int c0    = ctile * CT;

    // ---- stage per-pixel kernels: 1600 f32 = 800 b64, 25 exact iterations
    {
        const float* kb = kern + (size_t)b * (KSZ * KSZ) * HW + (size_t)h0 * IMW + w0;
        #pragma unroll 1
        for (int it = 0; it < 25; ++it) {
            int idx = lane + 32 * it;       // 0..799 (pair index)
            int j   = (idx & 7) * 2;
            int hr  = (idx >> 3) & 3;
            int kl  = idx >> 5;             // k*5 + l, 0..24
            v2f v = *(const v2f*)(kb + (size_t)kl * HW + (size_t)hr * IMW + j);
            *(v2f*)(&sK[((hr * (KSZ * KSZ) + kl) * WT) + j]) = v;
        }
    }
    // ---- stage features: 128 strips of 20 f32 = 4 exact iterations, b64 moves
    {
        const float* fb = feat + (size_t)(b * CH + c0) * HW;
        #pragma unroll 1
        for (int it = 0; it < 4; ++it) {
            int s  = lane + 32 * it;        // strip id 0..127
            int c  = s & 15;
            int r  = s >> 4;                // staged row 0..7
            int hh = h0 + r - PADR;
            bool rok = (hh >= 0) && (hh < IMH);
            int hc = hh < 0 ? 0 : (hh > IMH - 1 ? IMH - 1 : hh);
            const float* src = fb + (size_t)c * HW + (size_t)hc * IMW;
            float* dst = &sF[(r * CT + c) * MW];
            #pragma unroll
            for (int mm = 0; mm < MW / 2; ++mm) {
                int ww = w0 - PADR + 2 * mm;                 // even
                int wc = ww < 0 ? 0 : (ww > IMW - 2 ? IMW - 2 : ww);
                v2f v = *(const v2f*)(src + wc);             // 8B-aligned
                v2f o;
                o.x = (rok && (unsigned)ww       < (unsigned)IMW) ? v.x : 0.0f;
                o.y = (rok && (unsigned)(ww + 1) < (unsigned)IMW) ? v.y : 0.0f;
                *(v2f*)(dst + 2 * mm) = o;
            }
        }
    }
    __syncthreads();

    const int jn = lane & 15;   // A: channel row; B/C/D: pixel column
    const int hi = lane >> 4;

    v8f acc[HT];
    #pragma unroll
    for (int hr = 0; hr < HT; ++hr)
        acc[hr] = (v8f){0.f, 0.f, 0.f, 0.f, 0.f, 0.f, 0.f, 0.f};

    // r = staged feature row = hr + k; A-fragments for row r feed every (hr,k)
    // pair with hr+k==r, so 40 A-loads serve 100 WMMAs.
    #pragma unroll
    for (int r = 0; r < FR; ++r) {
        const float* fr = &sF[r * CT * MW];
        v2f A[5];
        #pragma unroll
        for (int q = 0; q < 5; ++q) {       // K = 20 -> 5 chunks of 4
            int mA = 4 * q + 2 * hi;
            A[q].x = fr[jn * MW + mA];      // conflict-free b64 LDS read
            A[q].y = fr[jn * MW + mA + 1];
        }
        #pragma unroll
        for (int k = 0; k < KSZ; ++k) {
            int hr = r - k;
            if (hr < 0 || hr >= HT) continue;   // compile-time after unroll
            const float* kr = &sK[(hr * (KSZ * KSZ) + k * KSZ) * WT];
            #pragma unroll
            for (int q = 0; q < 5; ++q) {
                // banded B: W[m,j] = ker[k, m-j, j] if 0 <= m-j < 5 else 0
                int l0  = 4 * q + 2 * hi - jn;
                int l1  = l0 + 1;
                int li0 = l0 < 0 ? 0 : (l0 > KSZ - 1 ? KSZ - 1 : l0);
                int li1 = l1 < 0 ? 0 : (l1 > KSZ - 1 ? KSZ - 1 : l1);
                float kv0 = kr[li0 * WT + jn];
                float kv1 = kr[li1 * WT + jn];
                v2f bb;
                bb.x = ((unsigned)l0 < (unsigned)KSZ) ? kv0 : 0.0f;
                bb.y = ((unsigned)l1 < (unsigned)KSZ) ? kv1 : 0.0f;

                acc[hr] = __builtin_amdgcn_wmma_f32_16x16x4_f32(
                    false, A[q], false, bb, (short)0, acc[hr], false, false);
            }
        }
    }

    // C/D layout: VGPR v -> channel v + 8*hi, lane&15 -> pixel column
    #pragma unroll
    for (int hr = 0; hr < HT; ++hr) {
        float* ob = out + ((size_t)(b * CH + c0) * IMH + (h0 + hr)) * IMW + w0 + jn;
        #pragma unroll
        for (int v = 0; v < 8; ++v) {
            int c = v + 8 * hi;
            ob[(size_t)c * HW] = acc[hr][v];
        }
    }
}

extern "C" void kernel_launch(void* const* d_in, const int* in_sizes, int n_in,
                              void* d_out, int out_size, void* d_ws, size_t ws_size,
                              hipStream_t stream)
{
    const float* kern = (const float*)d_in[0];   // (8, 5, 5, 256, 256) f32
    const float* feat = (const float*)d_in[1];   // (8, 128, 256, 256) f32
    float* out = (float*)d_out;                  // (8, 128, 256, 256) f32

    dim3 grid(IMW / WT, IMH / HT, BATCH * (CH / CT)); // 16 x 64 x 64
    dim3 block(32);                                   // one wave32 per block
    hipLaunchKernelGGL(svconv_wmma_kernel, grid, block, 0, stream, kern, feat, out);
}